// TemporalSelfAttention_10170482557430
// MI455X (gfx1250) — compile-verified
//
#include <hip/hip_runtime.h>
#include <math.h>

// ---------------- problem constants (from reference) ----------------
#define BATCH 2
#define BEV   128
#define NQ    (BEV*BEV)          // 16384
#define CCH   256                // C
#define NH    8                  // heads
#define NP    4                  // points
#define HD    (CCH/NH)           // 32 = head dim
#define MTOT  (BATCH*NQ)         // 32768 rows for all GEMMs

typedef __attribute__((ext_vector_type(2))) float v2f;
typedef __attribute__((ext_vector_type(8))) float v8f;

// ---------------- elementwise q = query + query_pos ----------------
__global__ void ew_add_kernel(const float* __restrict__ a,
                              const float* __restrict__ b,
                              float* __restrict__ o, int n4) {
  int i = blockIdx.x * blockDim.x + threadIdx.x;
  if (i < n4) {
    const float4* a4 = (const float4*)a;
    const float4* b4 = (const float4*)b;
    float4 x = a4[i], y = b4[i];
    float4 r;
    r.x = x.x + y.x; r.y = x.y + y.y; r.z = x.z + y.z; r.w = x.w + y.w;
    ((float4*)o)[i] = r;
  }
}

// ---------------- WMMA f32 GEMM:  Out[M,N] = A[M,K] @ W[K,N] + bias (+ addend) ----------------
// Each wave computes an (MT*16) x 16 output slab: MT stacked 16x16 WMMA tiles
// sharing one B-operand fetch per K-step (amortizes the strided W loads MT-fold).
//
// A operand (16x4, 2 VGPRs):  lanes 0-15 -> rows M=lane,    K = {k0, k0+1}
//                             lanes 16-31-> rows M=lane-16, K = {k0+2, k0+3}
// B operand (4x16, 2 VGPRs):  lanes 0-15 -> cols N=lane,    K = {k0, k0+1}
//                             lanes 16-31-> cols N=lane-16, K = {k0+2, k0+3}
// C/D (16x16 f32, 8 VGPRs):   VGPR r: lanes 0-15 row m0+r, lanes 16-31 row m0+8+r.
template<int MT>
__global__ void wmma_gemm_f32(const float* __restrict__ A,
                              const float* __restrict__ W,
                              const float* __restrict__ bias,
                              const float* __restrict__ addend,
                              float* __restrict__ Out,
                              int M, int N, int K) {
  const int lane = threadIdx.x & 31;
  const int wave = threadIdx.x >> 5;
  const int wavesPerBlock = blockDim.x >> 5;
  const int tilesN = N >> 4;
  const int tilesM = M / (16 * MT);
  const int totalTiles = tilesM * tilesN;

  int tile = blockIdx.x * wavesPerBlock + wave;
  if (tile >= totalTiles) return;          // wave-uniform: EXEC stays all-ones for WMMA

  const int tm = tile / tilesN;
  const int tn = tile - tm * tilesN;
  const int m0 = tm * (16 * MT);
  const int n0 = tn << 4;

  const int half = lane >> 4;              // 0: K pair {0,1}; 1: K pair {2,3}
  const int l15  = lane & 15;
  const int bcol = n0 + l15;

  const float* __restrict__ Arow = A + (size_t)(m0 + l15) * K;

  // warm L2/WGP$ for this wave's A rows (global_prefetch_b8)
  __builtin_prefetch(Arow, 0, 0);
#pragma unroll
  for (int s = 1; s < MT; ++s) __builtin_prefetch(Arow + (size_t)(16 * s) * K, 0, 0);

  v8f acc[MT];
#pragma unroll
  for (int s = 0; s < MT; ++s) acc[s] = (v8f){};

  for (int k0 = 0; k0 < K; k0 += 4) {
    const int ka = k0 + half * 2;
    v2f bb;
    bb.x = W[(size_t)ka * N + bcol];
    bb.y = W[(size_t)(ka + 1) * N + bcol];
#pragma unroll
    for (int s = 0; s < MT; ++s) {
      v2f a = *(const v2f*)(Arow + (size_t)(16 * s) * K + ka);  // contiguous 8B load
      acc[s] = __builtin_amdgcn_wmma_f32_16x16x4_f32(
          /*neg_a=*/false, a, /*neg_b=*/false, bb,
          /*c_mod=*/(short)0, acc[s], /*reuse_a=*/false, /*reuse_b=*/false);
    }
  }

  const float bval = bias ? bias[bcol] : 0.0f;
#pragma unroll
  for (int s = 0; s < MT; ++s) {
#pragma unroll
    for (int r = 0; r < 8; ++r) {
      const int row = m0 + 16 * s + r + half * 8;
      float v = acc[s][r] + bval;
      if (addend) v += addend[(size_t)row * N + bcol];
      Out[(size_t)row * N + bcol] = v;
    }
  }
}

// ---------------- deformable bilinear sampling + attention combine ----------------
// One wave per (b*NQ + q, h); lane = channel d in [0,32).
// tmp[b,q, h*HD + d] = sum_p softmax(logits)[p] * bilinear(v_grid_h, loc_p)[d]
__global__ void deform_sample_kernel(const float* __restrict__ v,      // (B,NQ,C) projected value
                                     const float* __restrict__ ref,    // (B,NQ,1,2)
                                     const float* __restrict__ off,    // (B,NQ, NH*NP*2)
                                     const float* __restrict__ logits, // (B,NQ, NH*NP)
                                     float* __restrict__ tmp)          // (B,NQ,C)
{
  const int lane = threadIdx.x & 31;
  const int wid  = blockIdx.x * (blockDim.x >> 5) + (threadIdx.x >> 5);
  const int total = MTOT * NH;
  if (wid >= total) return;

  const int h  = wid & (NH - 1);
  const int bq = wid >> 3;              // b*NQ + q
  const int b  = bq >> 14;              // / NQ (NQ = 16384)

  // reference point (wave-uniform loads)
  const float rx = ref[(size_t)bq * 2 + 0];
  const float ry = ref[(size_t)bq * 2 + 1];

  // softmax over the 4 taps of this head (redundant per lane; tiny)
  float lg[NP];
  float mx = -1e30f;
#pragma unroll
  for (int p = 0; p < NP; ++p) {
    lg[p] = logits[(size_t)bq * (NH * NP) + h * NP + p];
    mx = fmaxf(mx, lg[p]);
  }
  float ssum = 0.0f;
#pragma unroll
  for (int p = 0; p < NP; ++p) { lg[p] = expf(lg[p] - mx); ssum += lg[p]; }
  const float inv = 1.0f / ssum;

  const float* __restrict__ vb = v + (size_t)b * NQ * CCH + h * HD + lane; // channel slice
  float acc = 0.0f;

#pragma unroll
  for (int p = 0; p < NP; ++p) {
    const float ox = off[(size_t)bq * (NH * NP * 2) + h * (NP * 2) + p * 2 + 0];
    const float oy = off[(size_t)bq * (NH * NP * 2) + h * (NP * 2) + p * 2 + 1];
    // loc = ref + off/[W,H]; x = loc_x*W - 0.5
    const float x = rx * (float)BEV + ox - 0.5f;
    const float y = ry * (float)BEV + oy - 0.5f;
    const float x0f = floorf(x), y0f = floorf(y);
    const float dx = x - x0f, dy = y - y0f;
    const int x0 = (int)x0f, y0 = (int)y0f;

    const float w00 = (1.0f - dx) * (1.0f - dy);
    const float w10 = dx * (1.0f - dy);
    const float w01 = (1.0f - dx) * dy;
    const float w11 = dx * dy;

    float s = 0.0f;
    {
      const int ix = x0, iy = y0;
      if (ix >= 0 && ix < BEV && iy >= 0 && iy < BEV)
        s += w00 * vb[(size_t)(iy * BEV + ix) * CCH];
    }
    {
      const int ix = x0 + 1, iy = y0;
      if (ix >= 0 && ix < BEV && iy >= 0 && iy < BEV)
        s += w10 * vb[(size_t)(iy * BEV + ix) * CCH];
    }
    {
      const int ix = x0, iy = y0 + 1;
      if (ix >= 0 && ix < BEV && iy >= 0 && iy < BEV)
        s += w01 * vb[(size_t)(iy * BEV + ix) * CCH];
    }
    {
      const int ix = x0 + 1, iy = y0 + 1;
      if (ix >= 0 && ix < BEV && iy >= 0 && iy < BEV)
        s += w11 * vb[(size_t)(iy * BEV + ix) * CCH];
    }
    acc += lg[p] * inv * s;
  }

  tmp[(size_t)bq * CCH + h * HD + lane] = acc;
}

// ---------------- launch ----------------
extern "C" void kernel_launch(void* const* d_in, const int* in_sizes, int n_in,
                              void* d_out, int out_size, void* d_ws, size_t ws_size,
                              hipStream_t stream) {
  const float* query     = (const float*)d_in[0];
  const float* value     = (const float*)d_in[1];
  const float* identity  = (const float*)d_in[2];
  const float* query_pos = (const float*)d_in[3];
  const float* refpts    = (const float*)d_in[4];
  // d_in[5] spatial_shapes (int), d_in[6] level_start_index (int): constants here
  const float* Wv   = (const float*)d_in[7];
  const float* bv   = (const float*)d_in[8];
  const float* Woff = (const float*)d_in[9];
  const float* boff = (const float*)d_in[10];
  const float* Ww   = (const float*)d_in[11];
  const float* bw   = (const float*)d_in[12];
  const float* Wo   = (const float*)d_in[13];
  const float* bo   = (const float*)d_in[14];
  float* out = (float*)d_out;

  // workspace partition (all fp32)
  char* ws = (char*)d_ws;
  float* ws_q      = (float*)(ws);                                   // MTOT*256
  float* ws_v      = (float*)(ws + (size_t)MTOT * CCH * 4);          // MTOT*256
  float* ws_off    = (float*)(ws + (size_t)MTOT * CCH * 4 * 2);      // MTOT*64
  float* ws_logits = (float*)(ws + (size_t)MTOT * CCH * 4 * 2 +
                              (size_t)MTOT * 64 * 4);                // MTOT*32
  float* ws_tmp    = (float*)(ws + (size_t)MTOT * CCH * 4 * 2 +
                              (size_t)MTOT * 64 * 4 +
                              (size_t)MTOT * 32 * 4);                // MTOT*256

  const int WAVES_PER_BLOCK = 8;
  const int BLOCK = 32 * WAVES_PER_BLOCK;

  // 1) q = query + query_pos
  {
    int n4 = (MTOT * CCH) / 4;
    ew_add_kernel<<<(n4 + BLOCK - 1) / BLOCK, BLOCK, 0, stream>>>(query, query_pos, ws_q, n4);
  }

  // 2) v = value @ Wv + bv   (32768 x 256 x 256), 64-row slabs per wave
  {
    int tiles = (MTOT / 64) * (CCH / 16);
    wmma_gemm_f32<4><<<(tiles + WAVES_PER_BLOCK - 1) / WAVES_PER_BLOCK, BLOCK, 0, stream>>>(
        value, Wv, bv, nullptr, ws_v, MTOT, CCH, CCH);
  }

  // 3) off = q @ Woff + boff   (32768 x 256 x 64), 32-row slabs per wave
  {
    int tiles = (MTOT / 32) * (64 / 16);
    wmma_gemm_f32<2><<<(tiles + WAVES_PER_BLOCK - 1) / WAVES_PER_BLOCK, BLOCK, 0, stream>>>(
        ws_q, Woff, boff, nullptr, ws_off, MTOT, 64, CCH);
  }

  // 4) logits = q @ Ww + bw   (32768 x 256 x 32), 32-row slabs per wave
  {
    int tiles = (MTOT / 32) * (32 / 16);
    wmma_gemm_f32<2><<<(tiles + WAVES_PER_BLOCK - 1) / WAVES_PER_BLOCK, BLOCK, 0, stream>>>(
        ws_q, Ww, bw, nullptr, ws_logits, MTOT, 32, CCH);
  }

  // 5) bilinear sampling + attention combine -> ws_tmp (B,NQ,C)
  {
    int waves = MTOT * NH;  // 262144
    deform_sample_kernel<<<(waves + WAVES_PER_BLOCK - 1) / WAVES_PER_BLOCK, BLOCK, 0, stream>>>(
        ws_v, refpts, ws_off, ws_logits, ws_tmp);
  }

  // 6) out = ws_tmp @ Wo + bo + identity   (32768 x 256 x 256), 64-row slabs per wave
  {
    int tiles = (MTOT / 64) * (CCH / 16);
    wmma_gemm_f32<4><<<(tiles + WAVES_PER_BLOCK - 1) / WAVES_PER_BLOCK, BLOCK, 0, stream>>>(
        ws_tmp, Wo, bo, identity, out, MTOT, CCH, CCH);
  }
}